// TBBaselineModel_65652870087395
// MI455X (gfx1250) — compile-verified
//
#include <hip/hip_runtime.h>
#include <hip/hip_bf16.h>

typedef __attribute__((ext_vector_type(2))) float v2f;
typedef __attribute__((ext_vector_type(8))) float v8f;

#define FEATDIM 128  // FEAT == HID == 128 in the reference
#define BM      32   // M rows per block (divides 500000 and 200000)

// ---------------------------------------------------------------------------
// GEMM: C[M x 128] = A[M x 128] @ W[128 x 128] + bias[128]
// One block = 256 threads = 8 wave32s covering BM=32 rows x 128 cols.
// Each wave owns one 16-column slice and TWO stacked 16x16 output tiles that
// share every B fragment -> 1 global B load-pair per 2 WMMAs, halving W's
// L2 traffic vs a 16-row block. K walked in steps of 4 with
// V_WMMA_F32_16X16X4_F32 (exact fp32; the GEMM is HBM-bound on the A stream
// at ~22 us for 500k rows, so fp32 matrix throughput is not the limiter).
// Safe to run in place (C == A): each block stores only rows it staged into
// LDS behind a barrier before computing.
// ---------------------------------------------------------------------------
__global__ __launch_bounds__(256) void gemm128_wmma_f32(
    const float* __restrict__ A, const float* __restrict__ W,
    const float* __restrict__ bias, float* __restrict__ C, int M)
{
    __shared__ float sA[BM * FEATDIM];  // 16 KB

    const int tid  = threadIdx.x;
    const int row0 = blockIdx.x * BM;
    if (row0 >= M) return;

    // Cooperative load of the 32x128 A tile: 1024 float4s, 256 threads, 4 each.
    {
        const float4* src = (const float4*)(A + (size_t)row0 * FEATDIM);
        float4* dst = (float4*)sA;
        #pragma unroll
        for (int i = 0; i < 4; ++i)
            dst[tid + 256 * i] = src[tid + 256 * i];
    }
    __syncthreads();

    const int wave = tid >> 5;         // 0..7 -> N tile
    const int lane = tid & 31;
    const int half = lane >> 4;        // 0: lanes 0-15, 1: lanes 16-31
    const int l16  = lane & 15;
    const int n    = wave * 16 + l16;  // global column this lane owns in B/C/D

    v8f acc0 = {};  // rows row0 +  0..15 : VGPR v holds M = v + 8*half, N = l16
    v8f acc1 = {};  // rows row0 + 16..31

    #pragma unroll
    for (int k0 = 0; k0 < FEATDIM; k0 += 4) {
        const int ka = k0 + 2 * half;
        // B fragment (4x16), row-striped across lanes; shared by both M tiles.
        v2f b;
        b.x = W[(size_t)ka       * FEATDIM + n];
        b.y = W[(size_t)(ka + 1) * FEATDIM + n];
        // A fragments (16x4 each): lanes 0-15 carry K = k0+0,1; lanes 16-31 K = k0+2,3
        v2f a0, a1;
        a0.x = sA[l16 * FEATDIM + ka];
        a0.y = sA[l16 * FEATDIM + ka + 1];
        a1.x = sA[(16 + l16) * FEATDIM + ka];
        a1.y = sA[(16 + l16) * FEATDIM + ka + 1];
        acc0 = __builtin_amdgcn_wmma_f32_16x16x4_f32(
            false, a0, false, b, (short)0, acc0, false, false);
        acc1 = __builtin_amdgcn_wmma_f32_16x16x4_f32(
            false, a1, false, b, (short)0, acc1, false, false);
    }

    const float bn = bias[n];
    float* Crow = C + (size_t)row0 * FEATDIM;
    #pragma unroll
    for (int v = 0; v < 8; ++v) {
        const int m = v + 8 * half;
        Crow[(size_t)m * FEATDIM + n]        = acc0[v] + bn;
        Crow[(size_t)(m + 16) * FEATDIM + n] = acc1[v] + bn;
    }
}

// ---------------------------------------------------------------------------
__global__ void zero_f32(float* __restrict__ p, long long n)
{
    long long i      = (long long)blockIdx.x * blockDim.x + threadIdx.x;
    long long stride = (long long)gridDim.x * blockDim.x;
    for (; i < n; i += stride) p[i] = 0.0f;
}

// ---------------------------------------------------------------------------
// Scatter-mean accumulate: one wave32 per edge; each lane moves float4
// (16 B) -> one coalesced 512 B row per wave, 4 fp32 atomics per lane.
// ---------------------------------------------------------------------------
__global__ __launch_bounds__(256) void scatter_accum(
    const float* __restrict__ paper_h, const int* __restrict__ author_ids,
    const int* __restrict__ paper_ids, float* __restrict__ sums,
    float* __restrict__ counts, int num_edges)
{
    const int wid  = (int)(((long long)blockIdx.x * blockDim.x + threadIdx.x) >> 5);
    const int lane = threadIdx.x & 31;
    if (wid >= num_edges) return;

    const int aid = author_ids[wid];
    const int pid = paper_ids[wid];

    const float4 v = *(const float4*)(paper_h + (size_t)pid * FEATDIM + lane * 4);
    float* dst = sums + (size_t)aid * FEATDIM + lane * 4;
    atomicAdd(dst + 0, v.x);
    atomicAdd(dst + 1, v.y);
    atomicAdd(dst + 2, v.z);
    atomicAdd(dst + 3, v.w);
    if (lane == 0) atomicAdd(counts + aid, 1.0f);
}

// ---------------------------------------------------------------------------
// sums[i] /= max(counts[row], 1)  (row = i / 128), in place.
// ---------------------------------------------------------------------------
__global__ void normalize_rows(float* __restrict__ sums,
                               const float* __restrict__ counts, long long n)
{
    long long i = (long long)blockIdx.x * blockDim.x + threadIdx.x;
    if (i < n) {
        const float c = counts[i >> 7];
        sums[i] = sums[i] / fmaxf(c, 1.0f);
    }
}

// ---------------------------------------------------------------------------
// Edgewise dot product: one wave32 per supervision edge, float4 per lane,
// butterfly reduction across the 32 lanes.
// ---------------------------------------------------------------------------
__global__ __launch_bounds__(256) void edge_dot(
    const float* __restrict__ author_h, const float* __restrict__ paper_h,
    const int* __restrict__ lab_a, const int* __restrict__ lab_p,
    float* __restrict__ out, int num_edges)
{
    const int wid  = (int)(((long long)blockIdx.x * blockDim.x + threadIdx.x) >> 5);
    const int lane = threadIdx.x & 31;
    if (wid >= num_edges) return;

    const int a = lab_a[wid];
    const int p = lab_p[wid];
    const float4 va = *(const float4*)(author_h + (size_t)a * FEATDIM + lane * 4);
    const float4 vp = *(const float4*)(paper_h + (size_t)p * FEATDIM + lane * 4);
    float s = va.x * vp.x + va.y * vp.y + va.z * vp.z + va.w * vp.w;

    #pragma unroll
    for (int off = 16; off > 0; off >>= 1)
        s += __shfl_xor(s, off, 32);

    if (lane == 0) out[wid] = s;
}

// ---------------------------------------------------------------------------
extern "C" void kernel_launch(void* const* d_in, const int* in_sizes, int n_in,
                              void* d_out, int out_size, void* d_ws, size_t ws_size,
                              hipStream_t stream)
{
    const float* paper_x    = (const float*)d_in[0];
    const int*   author_ids = (const int*)  d_in[1];
    const int*   paper_ids  = (const int*)  d_in[2];
    const int*   lab_a      = (const int*)  d_in[3];
    const int*   lab_p      = (const int*)  d_in[4];
    const float* W_paper    = (const float*)d_in[5];
    const float* b_paper    = (const float*)d_in[6];
    const float* W_author   = (const float*)d_in[7];
    const float* b_author   = (const float*)d_in[8];

    const int num_papers      = in_sizes[0] / FEATDIM;   // 500000
    const int num_edges       = in_sizes[1];             // 2000000
    const int num_label_edges = in_sizes[3];             // 500000
    const int num_authors     = 200000;                  // segment count (reference constant)

    // Workspace layout (fp32): paper_h | author_buf | counts  (~359 MB total)
    float* paper_h    = (float*)d_ws;
    float* author_buf = paper_h + (size_t)num_papers * FEATDIM;
    float* counts     = author_buf + (size_t)num_authors * FEATDIM;

    // 1) paper_h = paper_x @ W_paper + b_paper  (WMMA f32)
    gemm128_wmma_f32<<<num_papers / BM, 256, 0, stream>>>(
        paper_x, W_paper, b_paper, paper_h, num_papers);

    // 2) zero author sums + counts (contiguous region) — every call, so graph
    //    replays are deterministic.
    const long long nz = (long long)num_authors * FEATDIM + num_authors;
    zero_f32<<<2048, 256, 0, stream>>>(author_buf, nz);

    // 3) scatter-add gathered paper embeddings into author sums (8 edges/block)
    {
        const long long threads = (long long)num_edges * 32;
        const int blocks = (int)((threads + 255) / 256);
        scatter_accum<<<blocks, 256, 0, stream>>>(
            paper_h, author_ids, paper_ids, author_buf, counts, num_edges);
    }

    // 4) mean: divide rows by counts (clamped at 1)
    {
        const long long n = (long long)num_authors * FEATDIM;
        const int blocks = (int)((n + 255) / 256);
        normalize_rows<<<blocks, 256, 0, stream>>>(author_buf, counts, n);
    }

    // 5) author_h = author_buf @ W_author + b_author (in place; per-block rows
    //    are staged to LDS before any store, so C == A is safe)
    gemm128_wmma_f32<<<num_authors / BM, 256, 0, stream>>>(
        author_buf, W_author, b_author, author_buf, num_authors);

    // 6) pred[e] = dot(author_h[lab_a[e]], paper_h[lab_p[e]])
    {
        const long long threads = (long long)num_label_edges * 32;
        const int blocks = (int)((threads + 255) / 256);
        edge_dot<<<blocks, 256, 0, stream>>>(
            author_buf, paper_h, lab_a, lab_p, (float*)d_out, num_label_edges);
    }
}